// Qwen3_5DecoderLayer_11802570129924
// MI455X (gfx1250) — compile-verified
//
#include <hip/hip_runtime.h>

typedef __attribute__((ext_vector_type(2))) float v2f;
typedef __attribute__((ext_vector_type(8))) float v8f;
typedef __attribute__((ext_vector_type(4))) unsigned int u32x4;
typedef __attribute__((ext_vector_type(8))) int i32x8;
typedef __attribute__((ext_vector_type(4))) int i32x4;
typedef __attribute__((address_space(3))) char as3char;

namespace {
constexpr int BATCH = 2;
constexpr int SEQ   = 4096;
constexpr int HEADS = 16;
constexpr int DKC   = 128;          // Dk
constexpr int DVC   = 128;          // Dv
constexpr int CH    = 64;           // chunk length
constexpr int NCH   = SEQ / CH;     // 64 chunks
constexpr int GRP   = 4;            // Dv split -> 4 groups of 32 columns
constexpr int DVG   = DVC / GRP;    // 32
constexpr float QSCALE = 0.08838834764831845f;  // 128^-0.5
}

__device__ __forceinline__ v8f wmma4(v2f a, v2f b, v8f c) {
  // V_WMMA_F32_16X16X4_F32 : D = A(16x4,f32) x B(4x16,f32) + C(16x16,f32)
  return __builtin_amdgcn_wmma_f32_16x16x4_f32(false, a, false, b, (short)0, c,
                                               false, false);
}

// A fragment: A[row0 + (lane&15)][k0 + 2*(lane>>4) + {0,1}]
__device__ __forceinline__ v2f frag_a(const float* Sm, int ld, int row0, int k0,
                                      int h2, int rl) {
  const float* p = Sm + (row0 + rl) * ld + k0 + 2 * h2;
  v2f a; a.x = p[0]; a.y = p[1]; return a;
}
// B fragment (row-major B): B[k0 + 2*h2 + {0,1}][col0 + (lane&15)]
__device__ __forceinline__ v2f frag_b(const float* Sm, int ld, int k0, int col0,
                                      int h2, int rl) {
  v2f b;
  b.x = Sm[(k0 + 2 * h2    ) * ld + col0 + rl];
  b.y = Sm[(k0 + 2 * h2 + 1) * ld + col0 + rl];
  return b;
}
// B fragment of a transposed row-major matrix: B[d][c] = Sm[c][d]
__device__ __forceinline__ v2f frag_bt(const float* Sm, int ld, int k0, int col0,
                                       int h2, int rl) {
  const float* p = Sm + (col0 + rl) * ld + k0 + 2 * h2;
  v2f b; b.x = p[0]; b.y = p[1]; return b;
}

// ---- Tensor Data Mover: 2-D f32 tile (rows x cols, row stride in elements)
// D# layout per CDNA5 ISA ch.8: group0 {count|lds_addr|global_addr|type=2},
// group1 {data_size=4B, tensor_dim0/1, tile_dim0/1, tensor_dim0_stride}.
__device__ __forceinline__ unsigned lds_off(const void* p) {
  return (unsigned)(size_t)(as3char*)p;
}

__device__ __forceinline__ void tdm_load_tile_f32(unsigned lds_byte,
                                                  const float* gptr, int rows,
                                                  int cols, int stride_elems) {
  unsigned long long ga = (unsigned long long)(size_t)gptr;
  u32x4 g0;
  g0[0] = 1u;                                   // count=1, user mode
  g0[1] = lds_byte;                             // lds_addr (bytes)
  g0[2] = (unsigned)(ga & 0xFFFFFFFFu);         // global_addr[31:0]
  g0[3] = (unsigned)((ga >> 32) & 0x1FFFFFFu)   // global_addr[56:32]
          | (2u << 30);                         // type=2 ("image")
  i32x8 g1;
  g1[0] = (int)(2u << 16);                      // data_size=2 -> 4 bytes
  g1[1] = (int)((unsigned)cols << 16);          // tensor_dim0[15:0]
  g1[2] = (int)(((unsigned)cols >> 16) | ((unsigned)rows << 16)); // d0 hi|d1 lo
  g1[3] = (int)(((unsigned)rows >> 16) | ((unsigned)cols << 16)); // d1 hi|tile_dim0
  g1[4] = (int)rows;                            // tile_dim1 (tile_dim2=0)
  g1[5] = (int)stride_elems;                    // tensor_dim0_stride[31:0]
  g1[6] = 0;                                    // stride hi | dim1_stride lo
  g1[7] = 0;
  i32x4 z4 = {0, 0, 0, 0};
  i32x8 z8 = {0, 0, 0, 0, 0, 0, 0, 0};
  // clang-23 / therock-10.0 toolchain: 6-arg form
  __builtin_amdgcn_tensor_load_to_lds(g0, g1, z4, z4, z8, 0);
}

__global__ __launch_bounds__(256)
void gdn_chunk_scan(const float* __restrict__ qg, const float* __restrict__ kg,
                    const float* __restrict__ vg, const float* __restrict__ gg,
                    const float* __restrict__ bg, float* __restrict__ outg) {
  __shared__ float s_q[2][CH * DKC];    // 64 KB  double-buffered q chunk
  __shared__ float s_k[2][CH * DKC];    // 64 KB  double-buffered k chunk
  __shared__ float s_v[2][CH * DVG];    // 16 KB  double-buffered v chunk -> v_new
  __shared__ float s_tmp[CH * DVG];     //  8 KB  U -> (v_beta - U)
  __shared__ float s_state[DKC * DVG];  // 16 KB  recurrent state
  __shared__ float s_M[CH * CH];        // 16 KB  strict_tril(k_b k^T * decay)
  __shared__ float s_T[CH * CH];        // 16 KB  (I+M)^-1
  __shared__ float s_attn[CH * CH];     // 16 KB  tril(q k^T * decay)
  __shared__ float s_red[512];
  __shared__ float s_invq[CH], s_invk[CH];
  __shared__ float s_graw[CH], s_gc[CH], s_beta[CH];
  __shared__ float s_bg[CH], s_eg[CH], s_kd[CH];

  const int tid  = threadIdx.x;
  const int wave = __builtin_amdgcn_readfirstlane((int)(threadIdx.x >> 5));
  const int lane = tid & 31;
  const int h2   = lane >> 4;   // half-wave select
  const int rl   = lane & 15;   // row/col within tile

  const int bid   = blockIdx.x;
  const int grp   = bid % GRP;
  const int head  = (bid / GRP) % HEADS;
  const int batch = bid / (GRP * HEADS);
  const int dv0   = grp * DVG;

  for (int i = tid; i < DKC * DVG; i += 256) s_state[i] = 0.f;

  // kick off chunk-0 DMA (wave 0 issues; TENSORcnt tracks)
  if (wave == 0) {
    const float* qs = qg + ((size_t)batch * SEQ * HEADS + head) * DKC;
    const float* ks = kg + ((size_t)batch * SEQ * HEADS + head) * DKC;
    const float* vs = vg + ((size_t)batch * SEQ * HEADS + head) * DVC + dv0;
    tdm_load_tile_f32(lds_off(s_q[0]), qs, CH, DKC, HEADS * DKC);
    tdm_load_tile_f32(lds_off(s_k[0]), ks, CH, DKC, HEADS * DKC);
    tdm_load_tile_f32(lds_off(s_v[0]), vs, CH, DVG, HEADS * DVC);
  }
  __syncthreads();

  for (int n = 0; n < NCH; ++n) {
    const int s_base = n * CH;
    const int buf = n & 1;
    float* sq = s_q[buf];
    float* sk = s_k[buf];
    float* sv = s_v[buf];

    // ---- wait for this chunk's DMA; make LDS visible to all waves ----
    if (wave == 0) __builtin_amdgcn_s_wait_tensorcnt(0);
    __syncthreads();

    // ---- prefetch next chunk into the other buffer while we compute ----
    if (wave == 0 && n + 1 < NCH) {
      const size_t base = ((size_t)batch * SEQ + (s_base + CH)) * HEADS + head;
      tdm_load_tile_f32(lds_off(s_q[buf ^ 1]), qg + base * DKC, CH, DKC,
                        HEADS * DKC);
      tdm_load_tile_f32(lds_off(s_k[buf ^ 1]), kg + base * DKC, CH, DKC,
                        HEADS * DKC);
      tdm_load_tile_f32(lds_off(s_v[buf ^ 1]), vg + base * DVC + dv0, CH, DVG,
                        HEADS * DVC);
    }

    // ---------------- phase 1: scalars + sum-of-squares ----------------
    if (wave < 2) {  // tid 0..63
      size_t gsc = ((size_t)batch * SEQ + s_base + tid) * HEADS + head;
      s_graw[tid] = gg[gsc];
      s_beta[tid] = bg[gsc];
    }
    {
      const int row = tid >> 2, seg = tid & 3;
      const float4* q4 = reinterpret_cast<const float4*>(sq + row * DKC + seg * 32);
      const float4* k4 = reinterpret_cast<const float4*>(sk + row * DKC + seg * 32);
      float ssq = 0.f, ssk = 0.f;
#pragma unroll
      for (int j = 0; j < 8; ++j) {
        float4 a = q4[j]; ssq += a.x * a.x + a.y * a.y + a.z * a.z + a.w * a.w;
        float4 b = k4[j]; ssk += b.x * b.x + b.y * b.y + b.z * b.z + b.w * b.w;
      }
      s_red[row * 4 + seg]       = ssq;
      s_red[256 + row * 4 + seg] = ssk;
    }
    __syncthreads();

    // ---------------- phase 2: norms + parallel prefix of g ----------------
    if (wave < 2) {
      const int j = tid;
      float sq2 = s_red[j * 4] + s_red[j * 4 + 1] + s_red[j * 4 + 2] +
                  s_red[j * 4 + 3];
      s_invq[j] = rsqrtf(sq2 + 1e-6f) * QSCALE;
      float sk2 = s_red[256 + j * 4] + s_red[256 + j * 4 + 1] +
                  s_red[256 + j * 4 + 2] + s_red[256 + j * 4 + 3];
      s_invk[j] = rsqrtf(sk2 + 1e-6f);
      float acc = 0.f;
      for (int i = 0; i <= j; ++i) acc += s_graw[i];
      s_gc[j] = acc;
    }
    __syncthreads();

    // ---------------- phase 3: derived scalars + row rescale ----------------
    if (wave < 2) {
      const float gl = s_gc[CH - 1], gc = s_gc[tid];
      s_bg[tid] = s_beta[tid] * __expf(gc);  // k_beta * exp(gc) row scale
      s_eg[tid] = __expf(gc);                // q * exp(gc) row scale
      s_kd[tid] = __expf(gl - gc);           // k * exp(g_last - gc) row scale
    }
    {
      const int row = tid >> 2, seg = tid & 3;
      const float iq = s_invq[row], ik = s_invk[row];
      float* qr = sq + row * DKC + seg * 32;
      float* kr = sk + row * DKC + seg * 32;
#pragma unroll
      for (int j = 0; j < 32; ++j) { qr[j] *= iq; kr[j] *= ik; }
    }
    __syncthreads();

    // ---------------- phase 4: M and attn via WMMA ----------------
    // 16 M tiles + 16 attn tiles over [64x64]; 2 of each per wave.
    for (int tt = 0; tt < 2; ++tt) {
      const int ti = wave * 2 + tt;                  // scalar (wave uniform)
      const int row0 = (ti >> 2) * 16, col0 = (ti & 3) * 16;
      const float gCc = s_gc[col0 + rl];             // hoisted: fixed per lane
      // --- M tile: strict_tril(k_beta k^T * decay) ---
      if (col0 <= row0) {
        const float bb = s_beta[row0 + rl];          // hoisted row scale
        v8f acc = {0.f, 0.f, 0.f, 0.f, 0.f, 0.f, 0.f, 0.f};
#pragma unroll 8
        for (int kk = 0; kk < DKC; kk += 4) {
          v2f a = frag_a(sk, DKC, row0, kk, h2, rl);
          a.x *= bb; a.y *= bb;
          v2f b = frag_bt(sk, DKC, kk, col0, h2, rl);  // B[d][c] = k[c][d]
          acc = wmma4(a, b, acc);
        }
#pragma unroll
        for (int i = 0; i < 8; ++i) {
          int R = row0 + i + 8 * h2, Cc = col0 + rl;
          float val = acc[i] * __expf(s_gc[R] - gCc);
          s_M[R * CH + Cc] = (R > Cc) ? val : 0.f;
        }
      } else {  // strictly-upper tile: zeros
#pragma unroll
        for (int i = 0; i < 8; ++i)
          s_M[(row0 + i + 8 * h2) * CH + col0 + rl] = 0.f;
      }
      // --- attn tile: tril(q k^T * decay) ---
      {
        v8f acc = {0.f, 0.f, 0.f, 0.f, 0.f, 0.f, 0.f, 0.f};
#pragma unroll 8
        for (int kk = 0; kk < DKC; kk += 4) {
          v2f a = frag_a(sq, DKC, row0, kk, h2, rl);
          v2f b = frag_bt(sk, DKC, kk, col0, h2, rl);
          acc = wmma4(a, b, acc);
        }
#pragma unroll
        for (int i = 0; i < 8; ++i) {
          int R = row0 + i + 8 * h2, Cc = col0 + rl;
          float val = acc[i] * __expf(s_gc[R] - gCc);
          s_attn[R * CH + Cc] = (R >= Cc) ? val : 0.f;
        }
      }
    }
    __syncthreads();

    // ---- phase 5: T = (I+M)^-1 (waves 0-1)  ||  U = (k_b e^gc)@state ----
    if (wave < 2) {
      const int j = tid;  // column of T
      for (int i = 0; i < j; ++i) s_T[i * CH + j] = 0.f;
      for (int i = j; i < CH; ++i) {
        float acc = (i == j) ? 1.f : 0.f;
        for (int l = j; l < i; ++l) acc -= s_M[i * CH + l] * s_T[l * CH + j];
        s_T[i * CH + j] = acc;
      }
    }
    {
      const int row0 = (wave >> 1) * 16, col0 = (wave & 1) * 16;  // 8 tiles
      const float sc = s_bg[row0 + rl];                           // hoisted
      v8f acc = {0.f, 0.f, 0.f, 0.f, 0.f, 0.f, 0.f, 0.f};
#pragma unroll 8
      for (int kk = 0; kk < DKC; kk += 4) {
        v2f a = frag_a(sk, DKC, row0, kk, h2, rl);
        a.x *= sc; a.y *= sc;
        v2f b = frag_b(s_state, DVG, kk, col0, h2, rl);
        acc = wmma4(a, b, acc);
      }
#pragma unroll
      for (int i = 0; i < 8; ++i)
        s_tmp[(row0 + i + 8 * h2) * DVG + col0 + rl] = acc[i];
    }
    __syncthreads();

    // ---------------- phase 6: tmp = v_beta - U ----------------
    for (int idx = tid; idx < CH * DVG; idx += 256) {
      int r = idx >> 5;  // idx / DVG
      s_tmp[idx] = sv[idx] * s_beta[r] - s_tmp[idx];
    }
    __syncthreads();

    // ---------------- phase 7: v_new = T @ tmp  (into sv) ----------------
    {
      const int row0 = (wave >> 1) * 16, col0 = (wave & 1) * 16;
      v8f acc = {0.f, 0.f, 0.f, 0.f, 0.f, 0.f, 0.f, 0.f};
#pragma unroll 8
      for (int kk = 0; kk < CH; kk += 4) {
        v2f a = frag_a(s_T, CH, row0, kk, h2, rl);
        v2f b = frag_b(s_tmp, DVG, kk, col0, h2, rl);
        acc = wmma4(a, b, acc);
      }
#pragma unroll
      for (int i = 0; i < 8; ++i)
        sv[(row0 + i + 8 * h2) * DVG + col0 + rl] = acc[i];
    }
    __syncthreads();

    // ---------- phase 8: out = (q e^gc)@state + attn@v_new -> global ----------
    {
      const int row0 = (wave >> 1) * 16, col0 = (wave & 1) * 16;
      const float sc = s_eg[row0 + rl];  // hoisted
      v8f acc = {0.f, 0.f, 0.f, 0.f, 0.f, 0.f, 0.f, 0.f};
#pragma unroll 8
      for (int kk = 0; kk < DKC; kk += 4) {
        v2f a = frag_a(sq, DKC, row0, kk, h2, rl);
        a.x *= sc; a.y *= sc;
        v2f b = frag_b(s_state, DVG, kk, col0, h2, rl);
        acc = wmma4(a, b, acc);
      }
#pragma unroll 8
      for (int kk = 0; kk < CH; kk += 4) {
        v2f a = frag_a(s_attn, CH, row0, kk, h2, rl);
        v2f b = frag_b(sv, DVG, kk, col0, h2, rl);
        acc = wmma4(a, b, acc);
      }
#pragma unroll
      for (int i = 0; i < 8; ++i) {
        int R = row0 + i + 8 * h2;
        size_t go =
            ((size_t)((size_t)batch * SEQ + s_base + R) * HEADS + head) * DVC +
            dv0 + col0 + rl;
        outg[go] = acc[i];
      }
    }
    __syncthreads();

    // ------- phase 9: state = state*e^{g_last} + kd^T @ v_new -------
    {
      const float egl = __expf(s_gc[CH - 1]);
      for (int tt = 0; tt < 2; ++tt) {
        const int t = wave * 2 + tt;  // 16 tiles over [128 x 32]
        const int dk0 = (t >> 1) * 16, col0 = (t & 1) * 16;
        v8f acc;
#pragma unroll
        for (int i = 0; i < 8; ++i)
          acc[i] = s_state[(dk0 + i + 8 * h2) * DVG + col0 + rl] * egl;
#pragma unroll 8
        for (int kk = 0; kk < CH; kk += 4) {
          // A[m][kk'] = k[kk'][dk0+m] * s_kd[kk']  (transposed, K-scaled)
          v2f a;
          a.x = sk[(kk + 2 * h2) * DKC + dk0 + rl] * s_kd[kk + 2 * h2];
          a.y = sk[(kk + 2 * h2 + 1) * DKC + dk0 + rl] * s_kd[kk + 2 * h2 + 1];
          v2f b = frag_b(sv, DVG, kk, col0, h2, rl);
          acc = wmma4(a, b, acc);
        }
#pragma unroll
        for (int i = 0; i < 8; ++i)
          s_state[(dk0 + i + 8 * h2) * DVG + col0 + rl] = acc[i];
      }
    }
    __syncthreads();
  }
}

extern "C" void kernel_launch(void* const* d_in, const int* in_sizes, int n_in,
                              void* d_out, int out_size, void* d_ws,
                              size_t ws_size, hipStream_t stream) {
  const float* q    = (const float*)d_in[0];
  const float* k    = (const float*)d_in[1];
  const float* v    = (const float*)d_in[2];
  const float* g    = (const float*)d_in[3];
  const float* beta = (const float*)d_in[4];
  float* out = (float*)d_out;
  dim3 grid(BATCH * HEADS * GRP);  // 128 workgroups
  gdn_chunk_scan<<<grid, 256, 0, stream>>>(q, k, v, g, beta, out);
}